// MeanShift_88673894793675
// MI455X (gfx1250) — compile-verified
//
#include <hip/hip_runtime.h>

// Mean-shift filtering for MI455X (gfx1250), fp32 throughout.
// Pairwise -2*y.x for each 16x16 tile of the NxN kernel matrix is one
// V_WMMA_F32_16X16X4_F32 (K=4, C=3 padded with zeros). ||y||^2 rides in the
// WMMA C operand (loop-invariant); ||x||^2 is folded into the exp argument
// via one FMA. Weight = v_exp_f32 of a v_med3-clamped value: 3 VALU + 1 TRANS
// per weight beyond the accumulation FMAs.

#define BATCH  2
#define CHAN   3
#define NPTS   9216          // 96*96
#define MTILES (NPTS / 16)   // 576 row tiles per batch
#define WAVES_PER_BLOCK 8
#define NUM_ITERS 5
// -1/(2*bw^2) * log2(e) with bw = 0.1  ->  -50 * 1.4426950408889634
#define NEG50_LOG2E (-72.13475204444817f)

typedef __attribute__((ext_vector_type(2))) float v2f;
typedef __attribute__((ext_vector_type(8))) float v8f;

// ------------- prep: points packed as (x0,x1,x2,||x||^2), y0 -------------
__global__ void ms_prep(const float* __restrict__ x,
                        float4* __restrict__ pts4,
                        float4* __restrict__ y0) {
    int i = blockIdx.x * blockDim.x + threadIdx.x;   // over BATCH*NPTS
    if (i >= BATCH * NPTS) return;
    int b = i / NPTS;
    int n = i - b * NPTS;
    const float* xb = x + (size_t)b * CHAN * NPTS;   // planar BCHW
    float c0 = xb[n];
    float c1 = xb[NPTS + n];
    float c2 = xb[2 * NPTS + n];
    float sq = c0 * c0 + c1 * c1 + c2 * c2;
    pts4[i] = make_float4(c0, c1, c2, sq);   // .w carries ||x||^2
    y0[i]   = make_float4(c0, c1, c2, 1.0f);
}

// ---------------- one mean-shift iteration ----------------
__global__ void __launch_bounds__(32 * WAVES_PER_BLOCK)
ms_iter(const float4* __restrict__ yin,
        const float4* __restrict__ pts4,
        float4* __restrict__ yout) {
    const int lane = threadIdx.x & 31;
    const int wave = threadIdx.x >> 5;
    const int tile = blockIdx.x * WAVES_PER_BLOCK + wave;   // 0..BATCH*MTILES-1
    const int b    = tile / MTILES;
    const int mt   = tile - b * MTILES;
    const int r    = lane & 15;   // tile column / A-fetch row
    const int hi   = lane >> 4;   // half-wave selector

    const float4* yb = yin  + b * NPTS;
    const float4* pb = pts4 + b * NPTS;

    // A operand: lane supplies A[m = lane%16][k = 2*hi + v], A = -2*y, k=3 -> 0
    const int row = mt * 16 + r;
    float4 yr = yb[row];
    float ysq_own = yr.x * yr.x + yr.y * yr.y + yr.z * yr.z;
    v2f a;
    a.x = hi ? (-2.0f * yr.z) : (-2.0f * yr.x);
    a.y = hi ? 0.0f           : (-2.0f * yr.y);

    // Loop-invariant C operand: c[v] = ||y_m||^2, m = v + 8*hi
    v8f c;
#pragma unroll
    for (int v = 0; v < 8; ++v)
        c[v] = __shfl(ysq_own, v + 8 * hi, 32);

    float accx[8], accy[8], accz[8], accw[8];
#pragma unroll
    for (int v = 0; v < 8; ++v) { accx[v] = 0.f; accy[v] = 0.f; accz[v] = 0.f; accw[v] = 0.f; }

#pragma unroll 2
    for (int nt = 0; nt < NPTS; nt += 16) {
        float4 p = pb[nt + r];               // one b128: point + ||x||^2
        float xqn = p.w * NEG50_LOG2E;       // fold ||x||^2 into exp argument

        // B operand: lane supplies B[k = 2*hi + v][n = lane%16], k=3 -> 0
        v2f bm;
        bm.x = hi ? p.z : p.x;
        bm.y = hi ? 0.0f : p.y;

        // d[v] = ||y_m||^2 - 2 y.x   (16x16 tile in one wmma)
        v8f d = __builtin_amdgcn_wmma_f32_16x16x4_f32(
            false, a, false, bm, (short)0, c, false, false);

#pragma unroll
        for (int v = 0; v < 8; ++v) {
            // t = d2 * (-50*log2e); clamp max(d2,0) == min(t,0) == med3(t,0,-inf)
            float t = fmaf(d[v], NEG50_LOG2E, xqn);
            t = __builtin_amdgcn_fmed3f(t, 0.0f, -__builtin_inff());
            float w = __builtin_amdgcn_exp2f(t);   // exp(-50*max(d2,0))
            accx[v] = fmaf(w, p.x, accx[v]);
            accy[v] = fmaf(w, p.y, accy[v]);
            accz[v] = fmaf(w, p.z, accz[v]);
            accw[v] += w;                          // denominator
        }
    }

    // Reduce across the 16 lanes of each half-wave (tile columns)
#pragma unroll
    for (int off = 1; off < 16; off <<= 1) {
#pragma unroll
        for (int v = 0; v < 8; ++v) {
            accx[v] += __shfl_xor(accx[v], off, 32);
            accy[v] += __shfl_xor(accy[v], off, 32);
            accz[v] += __shfl_xor(accz[v], off, 32);
            accw[v] += __shfl_xor(accw[v], off, 32);
        }
    }

    // Lanes r<8 of each half write one row each: m = r + 8*hi
    if (r < 8) {
        const int v = r;
        const int m = mt * 16 + v + 8 * hi;
        float inv = 1.0f / accw[v];
        yout[b * NPTS + m] = make_float4(accx[v] * inv, accy[v] * inv, accz[v] * inv, 1.0f);
    }
}

// ---------------- final: [B,N,4] -> planar [B,C,H,W] ----------------
__global__ void ms_final(const float4* __restrict__ y, float* __restrict__ out) {
    int i = blockIdx.x * blockDim.x + threadIdx.x;
    if (i >= BATCH * NPTS) return;
    int b = i / NPTS;
    int n = i - b * NPTS;
    float4 p = y[i];
    float* ob = out + (size_t)b * CHAN * NPTS;
    ob[n]            = p.x;
    ob[NPTS + n]     = p.y;
    ob[2 * NPTS + n] = p.z;
}

extern "C" void kernel_launch(void* const* d_in, const int* in_sizes, int n_in,
                              void* d_out, int out_size, void* d_ws, size_t ws_size,
                              hipStream_t stream) {
    (void)in_sizes; (void)n_in; (void)out_size; (void)ws_size;
    const float* x = (const float*)d_in[0];
    float* ws = (float*)d_ws;

    // workspace layout (floats): pts4 | y0 | y1   (~885 KB total)
    float4* pts4 = (float4*)ws;                                   // 4*B*N
    float4* y0   = (float4*)(ws + (size_t)4 * BATCH * NPTS);      // 4*B*N
    float4* y1   = (float4*)(ws + (size_t)8 * BATCH * NPTS);      // 4*B*N

    const int total = BATCH * NPTS;
    ms_prep<<<(total + 255) / 256, 256, 0, stream>>>(x, pts4, y0);

    float4* yi = y0;
    float4* yo = y1;
    const int iter_blocks = (BATCH * MTILES) / WAVES_PER_BLOCK;   // 1152/8 = 144, exact
    for (int it = 0; it < NUM_ITERS; ++it) {
        ms_iter<<<iter_blocks, 32 * WAVES_PER_BLOCK, 0, stream>>>(yi, pts4, yo);
        float4* t = yi; yi = yo; yo = t;
    }

    ms_final<<<(total + 255) / 256, 256, 0, stream>>>(yi, (float*)d_out);
}